// VITS_48730698941069
// MI455X (gfx1250) — compile-verified
//
#include <hip/hip_runtime.h>

#define Bn  16
#define Dn  192
#define TXn 512
#define TYn 2048
#define NEGC (-1e9f)
#define CH  8   // DP rows per TDM chunk (2 x 8 x 512 x 4B = 32 KB LDS)

typedef __attribute__((ext_vector_type(2))) float v2f;
typedef __attribute__((ext_vector_type(8))) float v8f;
typedef __attribute__((ext_vector_type(4))) unsigned int u32x4;
typedef __attribute__((ext_vector_type(4))) int i32x4;
typedef __attribute__((ext_vector_type(8))) int i32x8;

// ---------------------------------------------------------------------------
// f32 WMMA 16x16x4 (D = A(16x4) * B(4x16) + C), wave32.  (codegen-confirmed
// last round: emits v_wmma_f32_16x16x4_f32)
// ---------------------------------------------------------------------------
#if __has_builtin(__builtin_amdgcn_wmma_f32_16x16x4_f32)
__device__ __forceinline__ v8f wmma_f32_16x16x4(v2f a, v2f b, v8f c) {
  return __builtin_amdgcn_wmma_f32_16x16x4_f32(false, a, false, b, (short)0, c,
                                               false, false);
}
#else
__device__ __forceinline__ v8f wmma_f32_16x16x4(v2f a, v2f b, v8f c) {
#pragma unroll
  for (int i = 0; i < 8; ++i) c[i] += a[0] * b[0] + a[1] * b[1];
  return c;
}
#endif

// ---------------------------------------------------------------------------
// Tensor Data Mover: async 2D tile load global -> LDS, tracked by TENSORcnt.
// D# bitfields per CDNA5 ISA ch.8 (group0: count/lds_addr/global_addr/type,
// group1: data_size/tensor dims/tile dims/strides).  2D tensor -> groups 2/3
// zero (NULL-equivalent payload).
// ---------------------------------------------------------------------------
#if __has_builtin(__builtin_amdgcn_tensor_load_to_lds) && \
    __has_builtin(__builtin_amdgcn_s_wait_tensorcnt)
#define HAVE_TDM 1
__device__ __forceinline__ void vits_tdm_issue(const float* gbase_row,
                                               int rows_remaining,
                                               void* lds_dst) {
  unsigned long long ga = (unsigned long long)(size_t)gbase_row;
  u32x4 g0;
  g0[0] = 1u;                                   // count=1, user mode
  g0[1] = (unsigned int)(size_t)lds_dst;        // LDS byte address
  g0[2] = (unsigned int)(ga & 0xffffffffull);   // global_addr[31:0]
  g0[3] = (unsigned int)((ga >> 32) & 0x1ffffffull) | 0x80000000u; // addr[56:32] | type=2
  i32x8 g1;
  g1[0] = 0x20000;                              // data_size = 2 (4 bytes)
  g1[1] = (int)((unsigned)(TXn & 0xffff) << 16);            // tensor_dim0 lo16
  g1[2] = (int)(((unsigned)rows_remaining & 0xffffu) << 16); // dim0 hi16=0 | tensor_dim1 lo16
  g1[3] = (int)((unsigned)TXn << 16);           // tensor_dim1 hi16=0 | tile_dim0=512
  g1[4] = CH;                                   // tile_dim1=CH | tile_dim2=0
  g1[5] = TXn;                                  // tensor_dim0_stride lo32
  g1[6] = 0;                                    // stride0 hi16 | stride1 lo16
  g1[7] = 0;                                    // stride1 hi32
  i32x4 z4 = {0, 0, 0, 0};
#if __clang_major__ >= 23
  i32x8 z8 = {0, 0, 0, 0, 0, 0, 0, 0};
  __builtin_amdgcn_tensor_load_to_lds(g0, g1, z4, z4, z8, 0);
#else
  __builtin_amdgcn_tensor_load_to_lds(g0, g1, z4, z4, 0);
#endif
}
#endif

// ---------------------------------------------------------------------------
// Kernel 1: per-(b,x) precompute: s = exp(-2*logs), ms = mean*s, and
// nc14[b,x] = sum_d(-0.5*log(2pi) - logs - 0.5*mean^2*s).  Deterministic.
// ---------------------------------------------------------------------------
__global__ void vits_prep(const float* __restrict__ x_mean,
                          const float* __restrict__ x_logs,
                          float* __restrict__ msbuf,
                          float* __restrict__ sbuf,
                          float* __restrict__ nc14) {
  const int b = blockIdx.x >> 2;
  const int x = ((blockIdx.x & 3) << 7) + threadIdx.x;  // 128 threads
  float acc = 0.0f;
  for (int d = 0; d < Dn; ++d) {
    size_t idx = ((size_t)(b * Dn + d)) * TXn + x;
    float logs = x_logs[idx];
    float mean = x_mean[idx];
    float s    = __expf(-2.0f * logs);
    float ms   = mean * s;
    msbuf[idx] = ms;
    sbuf[idx]  = s;
    acc += (-0.91893853320467274178f - logs) - 0.5f * mean * ms;
  }
  nc14[b * TXn + x] = acc;
}

// ---------------------------------------------------------------------------
// Kernel 2: neg_cent via f32 WMMA.  One wave -> 4 M-tiles x 1 N-tile,
// K = 192 in steps of 4, two WMMAs per step (z*(mean*s) and (-0.5z^2)*s).
// ---------------------------------------------------------------------------
__global__ void vits_negcent_wmma(const float* __restrict__ z,
                                  const float* __restrict__ msbuf,
                                  const float* __restrict__ sbuf,
                                  const float* __restrict__ nc14,
                                  float* __restrict__ ncbuf) {
  const int lane  = threadIdx.x & 31;
  const int wv    = threadIdx.x >> 5;
  const int gtile = blockIdx.x * (blockDim.x >> 5) + wv;
  const int per_b = (TYn / 64) * (TXn / 16);
  const int b  = gtile / per_b;
  const int r  = gtile % per_b;
  const int y0 = (r / (TXn / 16)) * 64;
  const int x0 = (r % (TXn / 16)) * 16;

  const float* zb  = z     + (size_t)b * Dn * TYn;
  const float* msb = msbuf + (size_t)b * Dn * TXn;
  const float* sb  = sbuf  + (size_t)b * Dn * TXn;

  const int half = lane >> 4;
  const int l15  = lane & 15;

  v8f zero = {0, 0, 0, 0, 0, 0, 0, 0};
  v8f acc[4];
#pragma unroll
  for (int mt = 0; mt < 4; ++mt) acc[mt] = zero;

  for (int d0 = 0; d0 < Dn; d0 += 4) {
    const int k0 = d0 + (half << 1);
    float bm0 = msb[(size_t)k0 * TXn + x0 + l15];
    float bm1 = msb[(size_t)(k0 + 1) * TXn + x0 + l15];
    float bv0 = sb [(size_t)k0 * TXn + x0 + l15];
    float bv1 = sb [(size_t)(k0 + 1) * TXn + x0 + l15];
    v2f Bms = {bm0, bm1};
    v2f Bs  = {bv0, bv1};
#pragma unroll
    for (int mt = 0; mt < 4; ++mt) {
      int yy = y0 + (mt << 4) + l15;
      float a0 = zb[(size_t)k0 * TYn + yy];
      float a1 = zb[(size_t)(k0 + 1) * TYn + yy];
      v2f Az = {a0, a1};
      v2f Aq = {-0.5f * a0 * a0, -0.5f * a1 * a1};
      acc[mt] = wmma_f32_16x16x4(Az, Bms, acc[mt]);
      acc[mt] = wmma_f32_16x16x4(Aq, Bs,  acc[mt]);
    }
  }

  const float cadd = nc14[b * TXn + x0 + l15];
#pragma unroll
  for (int mt = 0; mt < 4; ++mt) {
#pragma unroll
    for (int rr = 0; rr < 8; ++rr) {
      int row = y0 + (mt << 4) + rr + (half << 3);
      ncbuf[((size_t)b * TYn + row) * TXn + x0 + l15] = acc[mt][rr] + cadd;
    }
  }
}

// ---------------------------------------------------------------------------
// MAS DP row update (shared by TDM and fallback paths).  Lane t owns columns
// [16t,16t+16) in registers; one __shfl_up per row for the chunk boundary.
// ---------------------------------------------------------------------------
__device__ __forceinline__ void mas_row_step(const float4* rp, int y, int t,
                                             float v[16],
                                             unsigned short* __restrict__ db) {
  float4 q0 = rp[0], q1 = rp[1], q2 = rp[2], q3 = rp[3];
  float rr[16] = {q0.x, q0.y, q0.z, q0.w, q1.x, q1.y, q1.z, q1.w,
                  q2.x, q2.y, q2.z, q2.w, q3.x, q3.y, q3.z, q3.w};
  float carry = __shfl_up(v[15], 1);
  if (t == 0) carry = NEGC;
  unsigned int bits = 0u;
#pragma unroll
  for (int i = 15; i >= 0; --i) {  // descending: v[i-1] still holds prev row
    float diag = (i == 0) ? carry : v[i - 1];
    unsigned int take = (diag > v[i]) ? 1u : 0u;  // recorded pre-mask (ref)
    float m  = take ? diag : v[i];
    float nv = rr[i] + m;
    int x = (t << 4) + i;
    v[i] = (x <= y) ? nv : NEGC;
    bits |= take << i;
  }
  db[(size_t)y * 32 + t] = (unsigned short)bits;
}

// ---------------------------------------------------------------------------
// Kernel 3: MAS forward DP.  One wave32 per batch.  With TDM: neg_cent rows
// are double-buffered into LDS by the Tensor Data Mover (chunk c+1 in flight
// while chunk c is consumed), so only ds_load latency sits on the serial
// chain.  Single wave per workgroup -> s_wait_tensorcnt is the only sync.
// ---------------------------------------------------------------------------
__global__ void vits_mas_forward(const float* __restrict__ ncbuf,
                                 unsigned short* __restrict__ dirs) {
  const int b = blockIdx.x;
  const int t = threadIdx.x;  // 0..31
  const float* nc = ncbuf + (size_t)b * TYn * TXn;
  unsigned short* db = dirs + (size_t)b * TYn * 32;

  float v[16];
#pragma unroll
  for (int i = 0; i < 16; ++i) v[i] = NEGC;
  if (t == 0) v[0] = nc[0];  // row 0: only x==0 is live

#if defined(HAVE_TDM)
  __shared__ __align__(16) float lbuf[2][CH][TXn];
  const int nchunk = (TYn - 1 + CH - 1) / CH;  // 256
  vits_tdm_issue(nc + (size_t)1 * TXn, TYn - 1, &lbuf[0][0][0]);
  int y = 1;
  for (int c = 0; c < nchunk; ++c) {
    if (c + 1 < nchunk) {
      const int ystart = 1 + CH * (c + 1);
      vits_tdm_issue(nc + (size_t)ystart * TXn, TYn - ystart,
                     &lbuf[(c + 1) & 1][0][0]);
      __builtin_amdgcn_s_wait_tensorcnt(1);  // chunk c complete (in-order)
    } else {
      __builtin_amdgcn_s_wait_tensorcnt(0);
    }
    const float* rows = &lbuf[c & 1][0][0];
    const int left = TYn - (1 + CH * c);
    const int ylim = (left < CH) ? left : CH;
    for (int j = 0; j < ylim; ++j, ++y) {
      const float4* rp =
          reinterpret_cast<const float4*>(rows + j * TXn + (t << 4));
      mas_row_step(rp, y, t, v, db);
    }
  }
#else
  for (int y = 1; y < TYn; ++y) {
    const float4* rp =
        reinterpret_cast<const float4*>(nc + (size_t)y * TXn + (t << 4));
    mas_row_step(rp, y, t, v, db);
  }
#endif
}

// ---------------------------------------------------------------------------
// Kernel 4: backtrack.  Wave loads full dirs rows (512 bits) 8 rows ahead
// (no dependent-load chain); serial chain resolved with __shfl broadcasts.
// ---------------------------------------------------------------------------
__global__ void vits_mas_backtrack(const unsigned short* __restrict__ dirs,
                                   int* __restrict__ path) {
  const int b    = blockIdx.x;
  const int lane = threadIdx.x;  // 32
  const unsigned short* db = dirs + (size_t)b * TYn * 32;
  int* pb = path + b * TYn;

  int idx = TXn - 1;
  for (int y = TYn - 1; y >= 1; y -= 8) {
    const int nch = (y >= 8) ? 8 : y;
    unsigned int w[8];
#pragma unroll
    for (int j = 0; j < 8; ++j)
      if (j < nch) w[j] = db[(size_t)(y - j) * 32 + lane];
    for (int j = 0; j < nch; ++j) {
      if (lane == 0) pb[y - j] = idx;
      unsigned int word = (unsigned int)__shfl((int)w[j], idx >> 4);
      idx -= (int)((word >> (idx & 15)) & 1u);
    }
  }
  if (lane == 0) pb[0] = idx;
}

// ---------------------------------------------------------------------------
// Kernel 5: attn one-hot write (covers every element -> no pre-zeroing).
// ---------------------------------------------------------------------------
__global__ void vits_attn_write(const int* __restrict__ path,
                                float* __restrict__ attn) {
  const int by = blockIdx.x;       // b*TY + y
  const int p  = path[by];
  const int xb = threadIdx.x << 2; // 128 threads x float4
  float4 v;
  v.x = (xb + 0 == p) ? 1.0f : 0.0f;
  v.y = (xb + 1 == p) ? 1.0f : 0.0f;
  v.z = (xb + 2 == p) ? 1.0f : 0.0f;
  v.w = (xb + 3 == p) ? 1.0f : 0.0f;
  reinterpret_cast<float4*>(attn + (size_t)by * TXn)[threadIdx.x] = v;
}

// ---------------------------------------------------------------------------
// Kernel 6: out_main gather  (m_p | logs_p)[b,c,y] = src[b, c%D, path[b,y]].
// ---------------------------------------------------------------------------
__global__ void vits_gather(const float* __restrict__ x_mean,
                            const float* __restrict__ x_logs,
                            const int* __restrict__ path,
                            float* __restrict__ out_main) {
  const int bc = blockIdx.x;  // b*(2D) + c
  const int b  = bc / (2 * Dn);
  const int c  = bc % (2 * Dn);
  const float* src = (c < Dn)
                         ? (x_mean + ((size_t)b * Dn + c) * TXn)
                         : (x_logs + ((size_t)b * Dn + (c - Dn)) * TXn);
  float* dst = out_main + (size_t)bc * TYn;
  const int* pb = path + b * TYn;
  for (int y = threadIdx.x; y < TYn; y += blockDim.x) dst[y] = src[pb[y]];
}

// ---------------------------------------------------------------------------
// Kernel 7: durations via binary search on the monotone path + l_length.
// ---------------------------------------------------------------------------
__global__ void vits_llength(const int* __restrict__ path,
                             const float* __restrict__ logw,
                             float* __restrict__ l_out) {
  __shared__ int   sp[TYn];
  __shared__ float red[512];
  const int b = blockIdx.x;
  for (int y = threadIdx.x; y < TYn; y += 512) sp[y] = path[b * TYn + y];
  __syncthreads();

  const int x = threadIdx.x;  // 512 threads == TXn
  int lo = 0, hi = TYn;
  while (lo < hi) { int mid = (lo + hi) >> 1; if (sp[mid] <  x) lo = mid + 1; else hi = mid; }
  const int lb = lo;
  lo = 0; hi = TYn;
  while (lo < hi) { int mid = (lo + hi) >> 1; if (sp[mid] <= x) lo = mid + 1; else hi = mid; }
  const float w = (float)(lo - lb);
  const float t = logw[b * TXn + x] - logf(w + 1e-6f);
  red[threadIdx.x] = t * t;
  __syncthreads();
  for (int s = 256; s > 0; s >>= 1) {
    if (threadIdx.x < s) red[threadIdx.x] += red[threadIdx.x + s];
    __syncthreads();
  }
  if (threadIdx.x == 0) l_out[b] = red[0];
}

// ---------------------------------------------------------------------------
extern "C" void kernel_launch(void* const* d_in, const int* in_sizes, int n_in,
                              void* d_out, int out_size, void* d_ws,
                              size_t ws_size, hipStream_t stream) {
  (void)in_sizes; (void)n_in; (void)out_size; (void)ws_size;
  const float* x_mean = (const float*)d_in[0];
  const float* x_logs = (const float*)d_in[1];
  const float* z      = (const float*)d_in[2];
  const float* logw   = (const float*)d_in[3];

  char* ws = (char*)d_ws;
  float* ncbuf = (float*)ws;  ws += (size_t)Bn * TYn * TXn * sizeof(float); // 64 MB
  float* msbuf = (float*)ws;  ws += (size_t)Bn * Dn * TXn * sizeof(float);  // 6 MB
  float* sbuf  = (float*)ws;  ws += (size_t)Bn * Dn * TXn * sizeof(float);  // 6 MB
  float* nc14  = (float*)ws;  ws += (size_t)Bn * TXn * sizeof(float);
  unsigned short* dirs = (unsigned short*)ws;
  ws += (size_t)Bn * TYn * 32 * sizeof(unsigned short);                     // 2 MB
  int* path = (int*)ws;                                                     // 128 KB

  float* out_main = (float*)d_out;                       // [B, 2D, TY]
  float* l_len    = out_main + (size_t)Bn * 2 * Dn * TYn;
  float* attn     = l_len + Bn;                          // [B, TY, TX]

  vits_prep<<<Bn * 4, 128, 0, stream>>>(x_mean, x_logs, msbuf, sbuf, nc14);

  const int waves = Bn * (TYn / 64) * (TXn / 16);        // 16384 waves
  vits_negcent_wmma<<<waves / 8, 256, 0, stream>>>(z, msbuf, sbuf, nc14, ncbuf);

  vits_mas_forward<<<Bn, 32, 0, stream>>>(ncbuf, dirs);
  vits_mas_backtrack<<<Bn, 32, 0, stream>>>(dirs, path);

  vits_attn_write<<<Bn * TYn, TXn / 4, 0, stream>>>(path, attn);
  vits_gather<<<Bn * 2 * Dn, 256, 0, stream>>>(x_mean, x_logs, path, out_main);
  vits_llength<<<Bn, 512, 0, stream>>>(path, logw, l_len);
}